// Attention_1288490189562
// MI455X (gfx1250) — compile-verified
//
#include <hip/hip_runtime.h>
#include <hip/hip_bf16.h>

// ---------------------------------------------------------------------------
// MI455X (gfx1250) multi-head attention forward, fp32 WMMA (16x16x4_f32),
// double-buffered LDS staging with async global->LDS copies when available.
// ---------------------------------------------------------------------------

typedef float v2f __attribute__((ext_vector_type(2)));
typedef float v8f __attribute__((ext_vector_type(8)));

#define SEQ   1024
#define CH    1024
#define NH    16
#define HD    64
#define BATCH 4

static constexpr float SCALE_F = 0.125f;   // 64^-0.5

// LDS strides (floats): 16B-aligned rows for b128 staging, bank-conflict-free
// WMMA fragment reads.
static constexpr int SROW = 1028;  // S rows
static constexpr int QSTR = 68;    // q tile
static constexpr int VSTR = 72;    // v tile
static constexpr int ASTR = 20;    // GEMM A tile
static constexpr int BSTR = 72;    // GEMM B tile

// ---- async global->LDS staging (gfx1250 ASYNCcnt path), with fallback -----
#if defined(__has_builtin)
#if __has_builtin(__builtin_amdgcn_global_load_async_to_lds_b128) && \
    __has_builtin(__builtin_amdgcn_s_wait_asynccnt)
#define HAVE_ASYNC_LDS 1
#endif
#endif

#if defined(HAVE_ASYNC_LDS)
// Builtin payload type is a GNU vector: int __attribute__((vector_size(16))).
typedef int v4i_gnu __attribute__((vector_size(4 * sizeof(int))));
typedef __attribute__((address_space(1))) v4i_gnu gv4i_t;  // global (device)
typedef __attribute__((address_space(3))) v4i_gnu lv4i_t;  // LDS (shared)
#endif

// Copy 16 contiguous bytes global -> LDS (per lane).
__device__ __forceinline__ void stage16(const float* g, float* l) {
#if defined(HAVE_ASYNC_LDS)
  __builtin_amdgcn_global_load_async_to_lds_b128((gv4i_t*)g, (lv4i_t*)l,
                                                 /*offset=*/0, /*cpol=*/0);
#else
  *(float4*)l = *(const float4*)g;
#endif
}

// Wait for this wave's outstanding async LDS copies (no-op in fallback).
__device__ __forceinline__ void async_join() {
#if defined(HAVE_ASYNC_LDS)
  __builtin_amdgcn_s_wait_asynccnt(0);
#endif
}

__device__ __forceinline__ v8f wmma_f32(v2f a, v2f b, v8f c) {
  // D = A(16x4) * B(4x16) + C(16x16), full fp32.
  return __builtin_amdgcn_wmma_f32_16x16x4_f32(
      /*neg_a=*/false, a, /*neg_b=*/false, b,
      /*c_mod=*/(short)0, c, /*reuse_a=*/false, /*reuse_b=*/false);
}

// ---------------------------------------------------------------------------
// Kernel 1/3: fp32 WMMA GEMM, block tile 64x64, BK=16, 4 waves (128 thr),
// double-buffered async LDS staging (2 barriers per K-step).
// MODE 0: out = A@B + bias, scattered into qkv layout [3][B][H][SEQ][HD]
// MODE 1: out = A@B + bias, plain row-major [M][CH]
// ---------------------------------------------------------------------------
template <int MODE>
__global__ __launch_bounds__(128) void gemm_wmma_kernel(
    const float* __restrict__ A, const float* __restrict__ Bm,
    const float* __restrict__ bias, float* __restrict__ out,
    int Kdim, int Ncols) {
  __shared__ float As[2][64 * ASTR];
  __shared__ float Bs[2][16 * BSTR];

  const int tid  = threadIdx.x;
  const int wave = tid >> 5;
  const int lane = tid & 31;
  const int half = lane >> 4;
  const int l16  = lane & 15;

  const int m0 = blockIdx.y * 64;
  const int n0 = blockIdx.x * 64;

  v8f acc[4] = {};

  const int a_row  = tid >> 1;            // 64 rows, 2 threads/row
  const int a_koff = (tid & 1) * 8;
  const int b_row  = tid >> 3;            // 16 rows, 8 threads/row
  const int b_coff = (tid & 7) * 8;
  const int mrow   = wave * 16 + l16;

  const float* aSrc = A + (size_t)(m0 + a_row) * Kdim + a_koff;
  const float* bSrc = Bm + (size_t)b_row * Ncols + n0 + b_coff;

  auto stage = [&](int k0, int buf) {
    const float* sa = aSrc + k0;
    stage16(sa, &As[buf][a_row * ASTR + a_koff]);
    stage16(sa + 4, &As[buf][a_row * ASTR + a_koff + 4]);
    const float* sb = bSrc + (size_t)k0 * Ncols;
    stage16(sb, &Bs[buf][b_row * BSTR + b_coff]);
    stage16(sb + 4, &Bs[buf][b_row * BSTR + b_coff + 4]);
  };

  stage(0, 0);
  int buf = 0;
  for (int k0 = 0; k0 < Kdim; k0 += 16, buf ^= 1) {
    async_join();          // staging of `buf` complete (this thread)
    __syncthreads();       // ... and visible to all; prev compute finished
    if (k0 + 16 < Kdim) stage(k0 + 16, buf ^ 1);

#pragma unroll
    for (int kk = 0; kk < 16; kk += 4) {
      // A-frag: lanes 0-15 -> (M=l16, K=kk,kk+1); lanes 16-31 -> K=kk+2,kk+3
      v2f a = *(const v2f*)&As[buf][mrow * ASTR + kk + 2 * half];
#pragma unroll
      for (int c = 0; c < 4; ++c) {
        v2f bf;
        bf.x = Bs[buf][(kk + 2 * half)     * BSTR + c * 16 + l16];
        bf.y = Bs[buf][(kk + 2 * half + 1) * BSTR + c * 16 + l16];
        acc[c] = wmma_f32(a, bf, acc[c]);
      }
    }
    __syncthreads();       // compute on `buf` done before it is re-staged
  }

  // Epilogue. C/D layout: VGPR j holds (M = j + 8*half, N = l16) per chunk.
  if (MODE == 0) {
#pragma unroll
    for (int c = 0; c < 4; ++c) {
      const int n3 = n0 + c * 16 + l16;          // [0, 3072)
      const float bv = bias[n3];
      const int which = n3 >> 10;                // 0=q 1=k 2=v
      const int cc = n3 & 1023;
      const int h = cc >> 6, d = cc & 63;
#pragma unroll
      for (int j = 0; j < 8; ++j) {
        const int m = m0 + wave * 16 + j + half * 8;   // [0, 4096)
        const int bb = m >> 10, n = m & 1023;
        out[((((size_t)which * BATCH + bb) * NH + h) * SEQ + n) * HD + d] =
            acc[c][j] + bv;
      }
    }
  } else {
#pragma unroll
    for (int c = 0; c < 4; ++c) {
      const int ncol = n0 + c * 16 + l16;
      const float bv = bias[ncol];
#pragma unroll
      for (int j = 0; j < 8; ++j) {
        const int m = m0 + wave * 16 + j + half * 8;
        out[(size_t)m * CH + ncol] = acc[c][j] + bv;
      }
    }
  }
}

// ---------------------------------------------------------------------------
// Kernel 2: attention core. One block = (b, h, 16 query rows), 128 threads.
// S = q k^T * scale -> exact softmax over full row in LDS -> post-softmax
// hard mask -> write attn -> O = P V. All matmuls via fp32 WMMA.
// ---------------------------------------------------------------------------
__global__ __launch_bounds__(128) void attn_core_kernel(
    const float* __restrict__ q, const float* __restrict__ k,
    const float* __restrict__ v, const int* __restrict__ mask,
    float* __restrict__ attn_out, float* __restrict__ ctx) {
  extern __shared__ float smem[];
  float* sS     = smem;                       // 16 * SROW
  float* qs     = sS + 16 * SROW;             // 16 * QSTR
  float* vs     = qs + 16 * QSTR;             // 2 * 16 * VSTR (double buffer)
  float* red    = vs + 2 * 16 * VSTR;         // 16 * 9
  float* rowmax = red + 16 * 9;               // 16
  float* rowsum = rowmax + 16;                // 16

  const int tid  = threadIdx.x;
  const int wave = tid >> 5;
  const int lane = tid & 31;
  const int half = lane >> 4;
  const int l16  = lane & 15;

  const int b = blockIdx.z, h = blockIdx.y, qt = blockIdx.x;
  const int qbase = qt * 16;

  // --- Stage q tile [16 x 64] into LDS -----------------------------------
  {
    const float* qp = q + (((size_t)b * NH + h) * SEQ + qbase) * HD;
    const int row = tid >> 3, off = (tid & 7) * 8;
    stage16(qp + row * HD + off, &qs[row * QSTR + off]);
    stage16(qp + row * HD + off + 4, &qs[row * QSTR + off + 4]);
  }
  async_join();
  __syncthreads();

  // --- Phase 1: S[16 x 1024] = q k^T * scale (waves split key tiles) -----
  {
    const float* kp = k + (((size_t)b * NH + h) * SEQ) * HD;
    for (int kt = wave; kt < SEQ / 16; kt += 4) {
      v8f acc = {};
      const float* kb = kp + (size_t)(kt * 16 + l16) * HD + 2 * half;
#pragma unroll
      for (int kk = 0; kk < HD; kk += 4) {
        v2f a  = *(const v2f*)&qs[l16 * QSTR + kk + 2 * half];
        v2f bf = *(const v2f*)(kb + kk);   // k[kt*16+l16][kk+2h], [kk+2h+1]
        acc = wmma_f32(a, bf, acc);
      }
#pragma unroll
      for (int j = 0; j < 8; ++j)
        sS[(j + half * 8) * SROW + kt * 16 + l16] = acc[j] * SCALE_F;
    }
  }
  __syncthreads();

  // --- Phase 2: exact softmax per row (8 threads per row) ----------------
  {
    const int r = tid >> 3, g = tid & 7;
    float* rowp = &sS[r * SROW + g * 128];
    float m = -3.0e38f;
    for (int i = 0; i < 128; ++i) m = fmaxf(m, rowp[i]);
    red[r * 9 + g] = m;
    __syncthreads();
    if (g == 0) {
      float mm = red[r * 9];
      for (int t = 1; t < 8; ++t) mm = fmaxf(mm, red[r * 9 + t]);
      rowmax[r] = mm;
    }
    __syncthreads();
    const float rm = rowmax[r];
    float s = 0.f;
    for (int i = 0; i < 128; ++i) {
      float e = __expf(rowp[i] - rm);
      rowp[i] = e;
      s += e;
    }
    red[r * 9 + g] = s;
    __syncthreads();
    if (g == 0) {
      float ss = red[r * 9];
      for (int t = 1; t < 8; ++t) ss += red[r * 9 + t];
      rowsum[r] = ss;       // strictly > 0
    }
    __syncthreads();
  }

  // --- Phase 3: normalize, post-softmax hard mask, write attn (coalesced)-
  {
    float* ab = attn_out + (((size_t)b * NH + h) * SEQ + qbase) * SEQ;
    const int* mb = mask + (size_t)qbase * SEQ;
#pragma unroll 4
    for (int it = 0; it < 128; ++it) {
      const int idx = it * 128 + tid;
      const int row = idx >> 10, col = idx & 1023;
      const float inv = 1.0f / rowsum[row];
      const float val =
          (mb[(size_t)row * SEQ + col] == 0) ? 0.f : sS[row * SROW + col] * inv;
      sS[row * SROW + col] = val;
      ab[(size_t)row * SEQ + col] = val;
    }
  }
  __syncthreads();

  // --- Phase 4: O[16 x 64] = P[16 x 1024] @ V[1024 x 64] -----------------
  // Wave w owns output columns d = w*16 .. w*16+15; V tiles double-buffered.
  {
    const float* vp = v + (((size_t)b * NH + h) * SEQ) * HD;
    const int srow = tid >> 3, soff = (tid & 7) * 8;

    auto stageV = [&](int kt, int bufi) {
      const float* src = vp + (size_t)(kt * 16 + srow) * HD + soff;
      float* dst = vs + bufi * 16 * VSTR + srow * VSTR + soff;
      stage16(src, dst);
      stage16(src + 4, dst + 4);
    };

    v8f o = {};
    stageV(0, 0);
    for (int kt = 0; kt < SEQ / 16; ++kt) {
      const int bufi = kt & 1;
      async_join();
      __syncthreads();
      if (kt + 1 < SEQ / 16) stageV(kt + 1, bufi ^ 1);

      const float* vb = vs + bufi * 16 * VSTR;
#pragma unroll
      for (int kk = 0; kk < 16; kk += 4) {
        v2f a = *(const v2f*)&sS[l16 * SROW + kt * 16 + kk + 2 * half];
        v2f bf;
        bf.x = vb[(kk + 2 * half)     * VSTR + wave * 16 + l16];
        bf.y = vb[(kk + 2 * half + 1) * VSTR + wave * 16 + l16];
        o = wmma_f32(a, bf, o);
      }
      __syncthreads();
    }
    // ctx layout [B][SEQ][CH] so proj GEMM reads it row-major.
    float* cb = ctx + ((size_t)b * SEQ + qbase) * CH + h * HD + wave * 16 + l16;
#pragma unroll
    for (int j = 0; j < 8; ++j) cb[(size_t)(j + half * 8) * CH] = o[j];
  }
}

// ---------------------------------------------------------------------------
// Host-side launch
// ---------------------------------------------------------------------------
extern "C" void kernel_launch(void* const* d_in, const int* in_sizes, int n_in,
                              void* d_out, int out_size, void* d_ws,
                              size_t ws_size, hipStream_t stream) {
  const float* x      = (const float*)d_in[0];  // [4,1024,1024]
  const int*   mask   = (const int*)  d_in[1];  // [1024,1024]
  const float* w_qkv  = (const float*)d_in[2];  // [1024,3072]
  const float* b_qkv  = (const float*)d_in[3];  // [3072]
  const float* w_proj = (const float*)d_in[4];  // [1024,1024]
  const float* b_proj = (const float*)d_in[5];  // [1024]

  float* out  = (float*)d_out;                              // [4,1024,1024]
  float* attn = out + (size_t)BATCH * SEQ * CH;             // [4,16,1024,1024]

  float* qkv = (float*)d_ws;                                // [3][B][H][SEQ][HD]
  float* ctx = qkv + (size_t)3 * BATCH * NH * SEQ * HD;     // [B][SEQ][CH]
  const float* q = qkv;
  const float* k = q + (size_t)BATCH * NH * SEQ * HD;
  const float* v = k + (size_t)BATCH * NH * SEQ * HD;

  // 1) QKV projection: [4096 x 1024] @ [1024 x 3072] + bias -> q/k/v layout
  {
    dim3 grid(3 * CH / 64, BATCH * SEQ / 64);
    gemm_wmma_kernel<0><<<grid, 128, 0, stream>>>(x, w_qkv, b_qkv, qkv, CH,
                                                  3 * CH);
  }

  // 2) attention core (one block per (b, h, 16 q-rows))
  {
    const size_t shmem =
        (size_t)(16 * SROW + 16 * QSTR + 2 * 16 * VSTR + 16 * 9 + 32) *
        sizeof(float);
    (void)hipFuncSetAttribute((const void*)attn_core_kernel,
                              hipFuncAttributeMaxDynamicSharedMemorySize,
                              (int)shmem);
    dim3 grid(SEQ / 16, NH, BATCH);
    attn_core_kernel<<<grid, 128, shmem, stream>>>(q, k, v, mask, attn, ctx);
  }

  // 3) output projection: [4096 x 1024] @ [1024 x 1024] + bias
  {
    dim3 grid(CH / 64, BATCH * SEQ / 64);
    gemm_wmma_kernel<1><<<grid, 128, 0, stream>>>(ctx, w_proj, b_proj, out, CH,
                                                  CH);
  }
}